// sinkhorn_28673201668464
// MI455X (gfx1250) — compile-verified
//
#include <hip/hip_runtime.h>
#include <math.h>

typedef __attribute__((ext_vector_type(2))) float v2f;
typedef __attribute__((ext_vector_type(8))) float v8f;

#define PN     4096
#define BATCH  2
#define THRESH 0.01f

// ---------------- workspace layout (in floats) ----------------
#define WS_XP   0        // BATCH*PN*4 : (x0,x1,x2,|x|^2)
#define WS_YP   32768    // BATCH*PN*4
#define WS_U    65536    // BATCH*PN
#define WS_V    73728    // BATCH*PN
#define WS_ERR  81920    // 3 slots
#define WS_DONE 81923    // int

__global__ void sk_init(float* u, float* v, float* err, int* done, float* out0) {
    int i = blockIdx.x * blockDim.x + threadIdx.x;
    if (i < BATCH * PN) { u[i] = 0.f; v[i] = 0.f; }
    if (i < 3) err[i] = 0.f;
    if (i == 0) { *done = 0; *out0 = 0.f; }
}

__global__ void sk_pack(const float* __restrict__ src, float4* __restrict__ dst, int n) {
    int i = blockIdx.x * blockDim.x + threadIdx.x;
    if (i < n) {
        float a = src[3 * i], b = src[3 * i + 1], c = src[3 * i + 2];
        dst[i] = make_float4(a, b, c, a * a + b * b + c * c);
    }
}

// Half-iteration: ua[i] = log_inv - LSE_j(M_ij) + ua[i],  M_ij = -C_ij + ua_i + vb_j
// Tile orientation: WMMA M-dim = j (reduced), N-dim = i (kept).
//   A[m=j, k] = (y0, y1, y2, vb_j - |y_j|^2)   (reloaded per chunk)
//   B[k, n=i] = (2x0, 2x1, 2x2, 1.0)           (constant per wave)
//   d[r] (lane q, half h) = 2*dot(y_j, x_i) + vb_j - |y_j|^2,  j = j0+r+8h, i = i0+q
// LSE over j runs along registers+chunks within each lane: no shuffles in the loop.
__global__ __launch_bounds__(64)
void sk_lse_update(const float4* __restrict__ ap, const float4* __restrict__ bp,
                   float* __restrict__ ua, const float* __restrict__ vb,
                   const int* __restrict__ done, float* __restrict__ err_acc,
                   float log_inv) {
    if (*done) return;
    const int wave = threadIdx.x >> 5;
    const int lane = threadIdx.x & 31;
    const int q    = lane & 15;
    const int half = lane >> 4;
    const int b    = blockIdx.x / (PN / 32);
    const int i0   = (blockIdx.x % (PN / 32)) * 32 + wave * 16;

    const float4* apb = ap + (size_t)b * PN;
    const float4* bpb = bp + (size_t)b * PN;
    float*        uab = ua + (size_t)b * PN;
    const float*  vbb = vb + (size_t)b * PN;

    // B fragment (x side, column n = q): lanes0-15 (K0,K1), lanes16-31 (K2,K3)
    float4 xa = apb[i0 + q];
    v2f bfrag;
    bfrag.x = half ? 2.f * xa.z : 2.f * xa.x;
    bfrag.y = half ? 1.f        : 2.f * xa.y;

    float uold = uab[i0 + q];
    float rowc = uold - xa.w;          // u_i - |x_i|^2

    float rm = -INFINITY, rs = 0.f;
    for (int j0 = 0; j0 < PN; j0 += 16) {
        float4 yb = bpb[j0 + q];
        float bias = vbb[j0 + q] - yb.w;
        v2f afrag;
        afrag.x = half ? yb.z : yb.x;
        afrag.y = half ? bias : yb.y;

        v8f d = {};
        d = __builtin_amdgcn_wmma_f32_16x16x4_f32(false, afrag, false, bfrag,
                                                  (short)0, d, false, false);
        float val[8];
#pragma unroll
        for (int r = 0; r < 8; ++r) val[r] = d[r] + rowc;

        float cm = val[0];
#pragma unroll
        for (int r = 1; r < 8; ++r) cm = fmaxf(cm, val[r]);
        float nm = fmaxf(rm, cm);
        float s = 0.f;
#pragma unroll
        for (int r = 0; r < 8; ++r) s += __expf(val[r] - nm);
        rs = rs * __expf(rm - nm) + s;
        rm = nm;
    }

    // merge the two j-halves (single cross-lane exchange)
    float om = __shfl_xor(rm, 16, 32);
    float os = __shfl_xor(rs, 16, 32);
    float nm = fmaxf(rm, om);
    float st = rs * __expf(rm - nm) + os * __expf(om - nm);
    float lse = nm + __logf(st);
    float un  = (log_inv - lse) + uold;          // EPS == 1
    if (half == 0) uab[i0 + q] = un;

    if (err_acc != nullptr) {
        float derr = fabsf(un - uold);
        derr += __shfl_xor(derr, 1, 16);
        derr += __shfl_xor(derr, 2, 16);
        derr += __shfl_xor(derr, 4, 16);
        derr += __shfl_xor(derr, 8, 16);
        if (lane == 0) atomicAdd(err_acc, derr);
    }
}

__global__ void sk_finalize_err(const float* err_acc, int* done) {
    if (threadIdx.x == 0 && blockIdx.x == 0) {
        if (*done == 0 && (*err_acc / (float)BATCH) < THRESH) *done = 1;
    }
}

// Final pass: pi = exp(M); distance += pi*C (if dist!=null); corr[i] = argmax_j M_ij
// Second WMMA broadcasts v_j into the same register layout: A2 = (0,0,0,v_j).
__global__ __launch_bounds__(64)
void sk_final(const float4* __restrict__ ap, const float4* __restrict__ bp,
              const float* __restrict__ ua, const float* __restrict__ vb,
              float* __restrict__ corr, float* __restrict__ dist) {
    const int wave = threadIdx.x >> 5;
    const int lane = threadIdx.x & 31;
    const int q    = lane & 15;
    const int half = lane >> 4;
    const int b    = blockIdx.x / (PN / 32);
    const int i0   = (blockIdx.x % (PN / 32)) * 32 + wave * 16;

    const float4* apb = ap + (size_t)b * PN;
    const float4* bpb = bp + (size_t)b * PN;
    const float*  uab = ua + (size_t)b * PN;
    const float*  vbb = vb + (size_t)b * PN;

    float4 xa = apb[i0 + q];
    v2f bfrag;
    bfrag.x = half ? 2.f * xa.z : 2.f * xa.x;
    bfrag.y = half ? 1.f        : 2.f * xa.y;

    float uold = uab[i0 + q];
    float rowc = uold - xa.w;

    float bm = -INFINITY;
    int   bi = 0;
    float acc = 0.f;
    for (int j0 = 0; j0 < PN; j0 += 16) {
        float4 yb = bpb[j0 + q];
        float vj  = vbb[j0 + q];
        float bias = vj - yb.w;
        v2f afrag, afrag2;
        afrag.x  = half ? yb.z : yb.x;
        afrag.y  = half ? bias : yb.y;
        afrag2.x = 0.f;
        afrag2.y = half ? vj : 0.f;

        v8f d = {};
        d = __builtin_amdgcn_wmma_f32_16x16x4_f32(false, afrag, false, bfrag,
                                                  (short)0, d, false, false);
        v8f d2 = {};
        d2 = __builtin_amdgcn_wmma_f32_16x16x4_f32(false, afrag2, false, bfrag,
                                                   (short)0, d2, false, false);
#pragma unroll
        for (int r = 0; r < 8; ++r) {
            float Mv = d[r] + rowc;
            float Cv = uold + d2[r] - Mv;        // C = u_i + v_j - M
            if (dist != nullptr) acc += __expf(Mv) * Cv;
            int j = j0 + r + 8 * half;
            if (Mv > bm) { bm = Mv; bi = j; }    // within-lane j is increasing: ties keep first
        }
    }

    // merge halves for argmax (lower j wins ties)
    float obm = __shfl_xor(bm, 16, 32);
    int   obi = __shfl_xor(bi, 16, 32);
    if (obm > bm || (obm == bm && obi < bi)) { bm = obm; bi = obi; }
    if (half == 0) corr[(size_t)b * PN + i0 + q] = (float)bi;

    if (dist != nullptr) {
        acc += __shfl_xor(acc, 1, 32);
        acc += __shfl_xor(acc, 2, 32);
        acc += __shfl_xor(acc, 4, 32);
        acc += __shfl_xor(acc, 8, 32);
        acc += __shfl_xor(acc, 16, 32);
        if (lane == 0) atomicAdd(dist, acc);
    }
}

extern "C" void kernel_launch(void* const* d_in, const int* in_sizes, int n_in,
                              void* d_out, int out_size, void* d_ws, size_t ws_size,
                              hipStream_t stream) {
    const float* x = (const float*)d_in[0];
    const float* y = (const float*)d_in[1];
    float* out = (float*)d_out;
    float* ws  = (float*)d_ws;

    float4* xp = (float4*)(ws + WS_XP);
    float4* yp = (float4*)(ws + WS_YP);
    float*  u  = ws + WS_U;
    float*  v  = ws + WS_V;
    float*  er = ws + WS_ERR;
    int*    dn = (int*)(ws + WS_DONE);

    const float log_inv = -8.317766166719343f;  // log(1/4096)
    const int npts = BATCH * PN;
    const int grid = BATCH * (PN / 32);         // 2 waves x 16 rows per block

    sk_init<<<(npts + 255) / 256, 256, 0, stream>>>(u, v, er, dn, out);
    sk_pack<<<(npts + 255) / 256, 256, 0, stream>>>(x, xp, npts);
    sk_pack<<<(npts + 255) / 256, 256, 0, stream>>>(y, yp, npts);

    // iter 0 (u) + err check
    sk_lse_update<<<grid, 64, 0, stream>>>(xp, yp, u, v, dn, er + 0, log_inv);
    sk_finalize_err<<<1, 32, 0, stream>>>(er + 0, dn);
    // iter 1 (v)
    sk_lse_update<<<grid, 64, 0, stream>>>(yp, xp, v, u, dn, nullptr, log_inv);
    // iter 2 (u) + err check
    sk_lse_update<<<grid, 64, 0, stream>>>(xp, yp, u, v, dn, er + 1, log_inv);
    sk_finalize_err<<<1, 32, 0, stream>>>(er + 1, dn);
    // iter 3 (v)
    sk_lse_update<<<grid, 64, 0, stream>>>(yp, xp, v, u, dn, nullptr, log_inv);
    // iter 4 (u)
    sk_lse_update<<<grid, 64, 0, stream>>>(xp, yp, u, v, dn, er + 2, log_inv);

    // outputs: [distance][corr1: B*PN][corr2: B*PN]
    sk_final<<<grid, 64, 0, stream>>>(xp, yp, u, v, out + 1, out);            // corr1 + dist
    sk_final<<<grid, 64, 0, stream>>>(yp, xp, v, u, out + 1 + npts, nullptr); // corr2
}